// GCN_14697378087275
// MI455X (gfx1250) — compile-verified
//
#include <hip/hip_runtime.h>

// GCN forward on MI455X (gfx1250, wave32).
// Roofline: edge scatter-add (3.2M edges x 64|32 f32 feats ~ 2.5 GB of
// gather+atomic traffic) dominates; dense GEMMs (<0.5 GFLOP) ride the
// exact-f32 WMMA 16x16x4 path with compile-time shapes (no divergence).

#define N_NODES  100000
#define N_EDGES  3200000
#define N_GRAPHS 1024
#define H1 64
#define H2 32
#define KPAD 8              // x/W1 K padded 6 -> 8
#define SLOPE 0.01f

typedef __attribute__((ext_vector_type(2))) float v2f;
typedef __attribute__((ext_vector_type(8))) float v8f;

// ---------------- packing (zero-pad K: 6 -> 8) ----------------

__global__ void pack_x_kernel(const float* __restrict__ x, float* __restrict__ xp) {
    int i = blockIdx.x * 256 + threadIdx.x;          // over N*8
    if (i >= N_NODES * KPAD) return;
    int n = i >> 3, k = i & 7;
    xp[i] = (k < 6) ? x[n * 6 + k] : 0.0f;
}

__global__ void pack_w1_kernel(const float* __restrict__ w, float* __restrict__ wp) {
    int i = blockIdx.x * 256 + threadIdx.x;          // over 8*64
    if (i >= KPAD * H1) return;
    int k = i >> 6, n = i & 63;
    wp[i] = (k < 6) ? w[k * H1 + n] : 0.0f;
}

// ---------------- degree / normalization ----------------

__global__ void init_deg_kernel(float* __restrict__ degf) {
    int n = blockIdx.x * 256 + threadIdx.x;
    if (n < N_NODES) degf[n] = 1.0f;                 // self-loop
}

__global__ void edge_deg_kernel(const int* __restrict__ dst, float* __restrict__ degf) {
    int e = blockIdx.x * 256 + threadIdx.x;
    if (e < N_EDGES) atomicAdd(&degf[dst[e]], 1.0f);
}

__global__ void make_dis_kernel(const float* __restrict__ degf, float* __restrict__ dis) {
    int n = blockIdx.x * 256 + threadIdx.x;
    if (n < N_NODES) dis[n] = rsqrtf(fmaxf(degf[n], 1.0f));
}

__global__ void zero_kernel(float* __restrict__ p, int n) {
    int i = blockIdx.x * 256 + threadIdx.x;
    if (i < n) p[i] = 0.0f;
}

// ---------------- dense GEMM via V_WMMA_F32_16X16X4_F32 ----------------
// D[M,OUT] = A[M,KD] @ B[KD,OUT], KD % 4 == 0.  One wave = one 16x16 tile.
// A 16x4:  VGPR j, lane l -> K = j + 2*(l>>4), M = l&15  (k pair contiguous
//          and even-offset => 8B-aligned b64 load)
// B 4x16:  VGPR j, lane l -> K = j + 2*(l>>4), N = l&15
// C/D:     VGPR j, lane l -> M = j + 8*(l>>4), N = l&15
template<int KD, int OUT>
__global__ void gemm_wmma_f32(const float* __restrict__ A,
                              const float* __restrict__ B,
                              float* __restrict__ D,
                              int total_tiles) {
    constexpr int NT = OUT / 16;
    int wid  = (blockIdx.x * blockDim.x + threadIdx.x) >> 5;
    int lane = threadIdx.x & 31;
    if (wid >= total_tiles) return;                  // wave-uniform guard

    int mt = wid / NT, nt = wid % NT;
    int ml   = lane & 15;
    int h    = lane >> 4;
    int m    = mt * 16 + ml;
    int ncol = nt * 16 + ml;

    const float* Arow = A + (size_t)m * KD + 2 * h;  // even offset: aligned v2f
    const float* Bcol = B + (size_t)(2 * h) * OUT + ncol;

    v8f c = {};
#pragma unroll
    for (int kb = 0; kb < KD; kb += 4) {
        v2f a = *(const v2f*)(Arow + kb);
        v2f b;
        b.x = Bcol[(kb + 0) * OUT];
        b.y = Bcol[(kb + 1) * OUT];
        c = __builtin_amdgcn_wmma_f32_16x16x4_f32(
                false, a, false, b, (short)0, c, false, false);
    }

    float* Dp = D + (size_t)(mt * 16 + 8 * h) * OUT + ncol;
#pragma unroll
    for (int j = 0; j < 8; ++j)
        Dp[j * OUT] = c[j];
}

// ---------------- edge scatter (the bandwidth-bound part) ----------------
// One thread = (edge, 4 consecutive features): float4 gather (b128, fully
// coalesced within an edge group) + 4 coalesced f32 atomics into agg[dst].
template<int F>
__global__ void scatter_kernel(const int* __restrict__ src,
                               const int* __restrict__ dst,
                               const float* __restrict__ dis,
                               const float* __restrict__ h,
                               float* __restrict__ agg) {
    constexpr int Q = F / 4;
    int i = blockIdx.x * 256 + threadIdx.x;          // over E*Q (<= 51.2M, fits int)
    if (i >= N_EDGES * Q) return;
    int e  = i / Q;
    int f4 = (i & (Q - 1)) * 4;
    int s = src[e], d = dst[e];
    float norm = dis[s] * dis[d];
    float4 hv = *(const float4*)(h + (size_t)s * F + f4);
    float* ap = agg + (size_t)d * F + f4;
    atomicAdd(ap + 0, hv.x * norm);
    atomicAdd(ap + 1, hv.y * norm);
    atomicAdd(ap + 2, hv.z * norm);
    atomicAdd(ap + 3, hv.w * norm);
}

// agg[n,f] = leaky_relu(agg[n,f] + hlin[n,f]*dis[n]^2 + bias[f])
// (hlin*dis^2 is the self-loop message, applied analytically)
template<int F>
__global__ void bias_act_kernel(const float* __restrict__ hlin,
                                const float* __restrict__ dis,
                                const float* __restrict__ bias,
                                float* __restrict__ agg) {
    constexpr int Q = F / 4;
    int i = blockIdx.x * 256 + threadIdx.x;          // over N*Q
    if (i >= N_NODES * Q) return;
    int n  = i / Q;
    int f4 = (i & (Q - 1)) * 4;
    float dn = dis[n];
    float d2 = dn * dn;
    float4 hv = *(const float4*)(hlin + (size_t)n * F + f4);
    float4 av = *(float4*)(agg + (size_t)n * F + f4);
    float4 bv = *(const float4*)(bias + f4);
    float4 r;
    r.x = av.x + hv.x * d2 + bv.x;  r.x = (r.x > 0.0f) ? r.x : SLOPE * r.x;
    r.y = av.y + hv.y * d2 + bv.y;  r.y = (r.y > 0.0f) ? r.y : SLOPE * r.y;
    r.z = av.z + hv.z * d2 + bv.z;  r.z = (r.z > 0.0f) ? r.z : SLOPE * r.z;
    r.w = av.w + hv.w * d2 + bv.w;  r.w = (r.w > 0.0f) ? r.w : SLOPE * r.w;
    *(float4*)(agg + (size_t)n * F + f4) = r;
}

// ---------------- pooling + head ----------------

__global__ void pool_kernel(const float* __restrict__ h2,
                            const int* __restrict__ batch,
                            float* __restrict__ sums,
                            float* __restrict__ cnt) {
    int i = blockIdx.x * 256 + threadIdx.x;
    if (i >= N_NODES * H2) return;
    int n = i >> 5, f = i & 31;
    int g = batch[n];
    atomicAdd(&sums[g * H2 + f], h2[i]);
    if (f == 0) atomicAdd(&cnt[g], 1.0f);
}

__global__ void final_kernel(const float* __restrict__ sums,
                             const float* __restrict__ cnt,
                             const float* __restrict__ Wlin,
                             const float* __restrict__ blin,
                             float* __restrict__ out) {
    int g = blockIdx.x * 256 + threadIdx.x;
    if (g >= N_GRAPHS) return;
    float inv = 1.0f / fmaxf(cnt[g], 1.0f);
    float o0 = blin[0], o1 = blin[1];
#pragma unroll
    for (int f = 0; f < H2; ++f) {
        float p = sums[g * H2 + f] * inv;
        o0 += p * Wlin[f * 2 + 0];
        o1 += p * Wlin[f * 2 + 1];
    }
    out[g * 2 + 0] = o0;
    out[g * 2 + 1] = o1;
}

// ---------------- launcher ----------------

extern "C" void kernel_launch(void* const* d_in, const int* in_sizes, int n_in,
                              void* d_out, int out_size, void* d_ws, size_t ws_size,
                              hipStream_t stream) {
    const float* x    = (const float*)d_in[0];
    const int*   ei   = (const int*)d_in[1];          // [2,E]: src row, dst row
    const int*   batch= (const int*)d_in[2];
    const float* W1   = (const float*)d_in[3];
    const float* b1   = (const float*)d_in[4];
    const float* W2   = (const float*)d_in[5];
    const float* b2   = (const float*)d_in[6];
    const float* Wlin = (const float*)d_in[7];
    const float* blin = (const float*)d_in[8];
    float* out = (float*)d_out;

    const int* src = ei;
    const int* dst = ei + N_EDGES;

    // workspace carve-up (all f32)
    float* dis  = (float*)d_ws;                        // N
    float* degf = dis + N_NODES;                       // N
    float* xp   = degf + N_NODES;                      // N*8   padded x
    float* w1p  = xp + (size_t)N_NODES * KPAD;         // 8*64  padded W1
    float* bufA = w1p + KPAD * H1;                     // N*64  agg1 -> h1
    float* bufB = bufA + (size_t)N_NODES * H1;         // N*64  h_lin1; then h_lin2|agg2
    float* sums = bufB + (size_t)N_NODES * H1;         // G*32
    float* cnt  = sums + N_GRAPHS * H2;                // G
    float* hlin2 = bufB;                               // N*32
    float* agg2  = bufB + (size_t)N_NODES * H2;        // N*32

    const int T = 256;
    int nb_nodes = (N_NODES + T - 1) / T;
    int nb_edges = (N_EDGES + T - 1) / T;

    // pack + degree + normalization
    pack_x_kernel<<<(N_NODES * KPAD + T - 1) / T, T, 0, stream>>>(x, xp);
    pack_w1_kernel<<<(KPAD * H1 + T - 1) / T, T, 0, stream>>>(W1, w1p);
    init_deg_kernel<<<nb_nodes, T, 0, stream>>>(degf);
    edge_deg_kernel<<<nb_edges, T, 0, stream>>>(dst, degf);
    make_dis_kernel<<<nb_nodes, T, 0, stream>>>(degf, dis);

    // layer 1: h_lin1 = xp @ W1p  (6250 m-tiles x 4 n-tiles, K=8)
    {
        int tiles = (N_NODES / 16) * (H1 / 16);
        gemm_wmma_f32<KPAD, H1><<<(tiles + 7) / 8, T, 0, stream>>>(xp, w1p, bufB, tiles);
    }
    zero_kernel<<<(N_NODES * H1 + T - 1) / T, T, 0, stream>>>(bufA, N_NODES * H1);
    {
        int total = N_EDGES * (H1 / 4);
        scatter_kernel<H1><<<(total + T - 1) / T, T, 0, stream>>>(src, dst, dis, bufB, bufA);
    }
    bias_act_kernel<H1><<<(N_NODES * (H1 / 4) + T - 1) / T, T, 0, stream>>>(bufB, dis, b1, bufA);
    // bufA now holds h1 [N,64]

    // layer 2: h_lin2 = h1 @ W2  (K=64, 2 n-tiles)
    {
        int tiles = (N_NODES / 16) * (H2 / 16);
        gemm_wmma_f32<H1, H2><<<(tiles + 7) / 8, T, 0, stream>>>(bufA, W2, hlin2, tiles);
    }
    zero_kernel<<<(N_NODES * H2 + T - 1) / T, T, 0, stream>>>(agg2, N_NODES * H2);
    {
        int total = N_EDGES * (H2 / 4);
        scatter_kernel<H2><<<(total + T - 1) / T, T, 0, stream>>>(src, dst, dis, hlin2, agg2);
    }
    bias_act_kernel<H2><<<(N_NODES * (H2 / 4) + T - 1) / T, T, 0, stream>>>(hlin2, dis, b2, agg2);
    // agg2 now holds h2 [N,32]

    // pooling + head
    zero_kernel<<<(N_GRAPHS * (H2 + 1) + T - 1) / T, T, 0, stream>>>(sums, N_GRAPHS * (H2 + 1));
    pool_kernel<<<(N_NODES * H2 + T - 1) / T, T, 0, stream>>>(agg2, batch, sums, cnt);
    final_kernel<<<(N_GRAPHS + T - 1) / T, T, 0, stream>>>(sums, cnt, Wlin, blin, out);
}